// DecoderRNN_6846177870457
// MI455X (gfx1250) — compile-verified
//
#include <hip/hip_runtime.h>

#define HDIM 1024
#define VDIM 50257

typedef __attribute__((ext_vector_type(2))) float v2f;
typedef __attribute__((ext_vector_type(8))) float v8f;

__device__ __forceinline__ float dot4(float4 a, float4 b) {
    return a.x * b.x + a.y * b.y + a.z * b.z + a.w * b.w;
}

// ---------------------------------------------------------------------------
// Kernel 1: embedding+ReLU fused into a single-step LSTM cell.
// One wave32 per hidden index k (1024 waves). Each wave computes the 4 gate
// dot-products against W_ih (with x) and W_hh (with h0), reduces across the
// wave, applies activations, writes h1/c1.
// ---------------------------------------------------------------------------
__global__ __launch_bounds__(256) void lstm_step_kernel(
    const int* __restrict__ ids,
    const float* __restrict__ h0,
    const float* __restrict__ c0,
    const float* __restrict__ emb,
    const float* __restrict__ Wih,
    const float* __restrict__ Whh,
    const float* __restrict__ bih,
    const float* __restrict__ bhh,
    float* __restrict__ out,      // out[V + k] = h1[k], out[V + H + k] = c1[k]
    float* __restrict__ h1_ws)    // 8B-aligned copy of h1 for kernel 2
{
    const int lane = threadIdx.x & 31;
    const int wave = threadIdx.x >> 5;
    const int k = blockIdx.x * 8 + wave;          // 0..1023

    const float4* x4   = (const float4*)(emb + (size_t)ids[0] * HDIM);
    const float4* h4   = (const float4*)h0;
    const float4* wi_i = (const float4*)(Wih + (size_t)k * HDIM);
    const float4* wi_f = (const float4*)(Wih + (size_t)(HDIM + k) * HDIM);
    const float4* wi_g = (const float4*)(Wih + (size_t)(2 * HDIM + k) * HDIM);
    const float4* wi_o = (const float4*)(Wih + (size_t)(3 * HDIM + k) * HDIM);
    const float4* wh_i = (const float4*)(Whh + (size_t)k * HDIM);
    const float4* wh_f = (const float4*)(Whh + (size_t)(HDIM + k) * HDIM);
    const float4* wh_g = (const float4*)(Whh + (size_t)(2 * HDIM + k) * HDIM);
    const float4* wh_o = (const float4*)(Whh + (size_t)(3 * HDIM + k) * HDIM);

    float si = 0.f, sf = 0.f, sg = 0.f, so = 0.f;
    #pragma unroll
    for (int j = lane; j < HDIM / 4; j += 32) {
        float4 xv = x4[j];
        xv.x = fmaxf(xv.x, 0.f); xv.y = fmaxf(xv.y, 0.f);
        xv.z = fmaxf(xv.z, 0.f); xv.w = fmaxf(xv.w, 0.f);   // ReLU(embedding)
        float4 hv = h4[j];
        si += dot4(xv, wi_i[j]) + dot4(hv, wh_i[j]);
        sf += dot4(xv, wi_f[j]) + dot4(hv, wh_f[j]);
        sg += dot4(xv, wi_g[j]) + dot4(hv, wh_g[j]);
        so += dot4(xv, wi_o[j]) + dot4(hv, wh_o[j]);
    }

    // wave32 reduction
    #pragma unroll
    for (int off = 16; off > 0; off >>= 1) {
        si += __shfl_xor(si, off, 32);
        sf += __shfl_xor(sf, off, 32);
        sg += __shfl_xor(sg, off, 32);
        so += __shfl_xor(so, off, 32);
    }

    if (lane == 0) {
        float zi = si + bih[k]            + bhh[k];
        float zf = sf + bih[HDIM + k]     + bhh[HDIM + k];
        float zg = sg + bih[2 * HDIM + k] + bhh[2 * HDIM + k];
        float zo = so + bih[3 * HDIM + k] + bhh[3 * HDIM + k];
        float ig = 1.f / (1.f + __expf(-zi));
        float fg = 1.f / (1.f + __expf(-zf));
        float gg = tanhf(zg);
        float og = 1.f / (1.f + __expf(-zo));
        float c1 = fg * c0[k] + ig * gg;
        float h1 = og * tanhf(c1);
        out[VDIM + k]        = h1;
        out[VDIM + HDIM + k] = c1;
        h1_ws[k] = h1;
    }
}

// ---------------------------------------------------------------------------
// Kernel 2: logits = h1 @ W_out.T + b_out via V_WMMA_F32_16X16X4_F32.
// One wave per 16-row tile of W_out. Per step (K chunk of 4):
//   A (16x4): h1[kb..kb+3] replicated over M. ISA A layout:
//       lanes 0-15 hold {K=0,K=1}, lanes 16-31 hold {K=2,K=3}.
//   B (4x16): B[k][n] = W_out[n0+n][kb+k]. ISA B layout (row striped across
//       lanes within a VGPR): lane n (0-15) VGPRs = {K=0,K=1}, lane n+16 =
//       {K=2,K=3}  ->  each lane does one 8B load of 2 consecutive floats
//       from its own W_out row; each 128B line is fully consumed over 8 steps.
// D row M=0 sits in acc[0] of lanes 0-15.
// ---------------------------------------------------------------------------
__global__ __launch_bounds__(256) void logits_wmma_kernel(
    const float* __restrict__ Wout,
    const float* __restrict__ bout,
    const float* __restrict__ h1,     // 8B-aligned (workspace)
    float* __restrict__ logits)
{
    const int lane = threadIdx.x & 31;
    const int wave = threadIdx.x >> 5;
    const int tile = blockIdx.x * 8 + wave;
    const int n0   = tile * 16;

    const int half = lane >> 4;               // 0: K={0,1}, 1: K={2,3}
    int row = n0 + (lane & 15);
    if (row > VDIM - 1) row = VDIM - 1;       // clamp tail loads in-bounds

    const float* wrow = Wout + (size_t)row * HDIM + 2 * half;
    const float* hp   = h1 + 2 * half;

    v8f acc0 = {}; v8f acc1 = {};             // two chains for WMMA ILP
    #pragma unroll 8
    for (int kb = 0; kb < HDIM; kb += 8) {
        v2f a0 = *(const v2f*)(hp + kb);
        v2f b0 = *(const v2f*)(wrow + kb);
        acc0 = __builtin_amdgcn_wmma_f32_16x16x4_f32(
            false, a0, false, b0, (short)0, acc0, false, false);
        v2f a1 = *(const v2f*)(hp + kb + 4);
        v2f b1 = *(const v2f*)(wrow + kb + 4);
        acc1 = __builtin_amdgcn_wmma_f32_16x16x4_f32(
            false, a1, false, b1, (short)0, acc1, false, false);
    }

    float r = acc0[0] + acc1[0];              // D[M=0][N=lane], lanes 0-15
    int n = n0 + lane;
    if (lane < 16 && n < VDIM) {
        logits[n] = r + bout[n];
    }
}

extern "C" void kernel_launch(void* const* d_in, const int* in_sizes, int n_in,
                              void* d_out, int out_size, void* d_ws, size_t ws_size,
                              hipStream_t stream) {
    (void)in_sizes; (void)n_in; (void)out_size; (void)ws_size;
    const int*   ids  = (const int*)  d_in[0];
    const float* h0   = (const float*)d_in[1];
    const float* c0   = (const float*)d_in[2];
    const float* emb  = (const float*)d_in[3];
    const float* Wih  = (const float*)d_in[4];
    const float* Whh  = (const float*)d_in[5];
    const float* bih  = (const float*)d_in[6];
    const float* bhh  = (const float*)d_in[7];
    const float* Wout = (const float*)d_in[8];
    const float* bout = (const float*)d_in[9];
    float* out  = (float*)d_out;              // [logits(V) | h1(H) | c1(H)]
    float* h1ws = (float*)d_ws;               // aligned h1 staging (4 KB)

    // Kernel 1: 1024 waves (one per hidden unit), 8 waves per block.
    lstm_step_kernel<<<HDIM / 8, 256, 0, stream>>>(
        ids, h0, c0, emb, Wih, Whh, bih, bhh, out, h1ws);

    // Kernel 2: one wave per 16-row tile of W_out.
    const int tiles = (VDIM + 15) / 16;       // 3142
    const int wgs   = (tiles + 7) / 8;        // 393
    logits_wmma_kernel<<<wgs, 256, 0, stream>>>(Wout, bout, h1ws, out);
}